// MambaEncoder_11476152615152
// MI455X (gfx1250) — compile-verified
//
#include <hip/hip_runtime.h>
#include <hip/hip_bf16.h>
#include <math.h>

// ---------------- model constants ----------------
#define NUM_LAYERS 6
#define D_MODEL    384
#define D_STATE    64
#define D_CONV     4
#define D_INNER    768
#define HEADDIM    96
#define NHEADS     8
#define CONV_DIM   896           // D_INNER + 2*D_STATE
#define D_IN_PROJ  1672          // 2*D_INNER + 2*D_STATE + NHEADS
#define DT_OFF     1664          // D_INNER + CONV_DIM
#define BATCH      8
#define SEQLEN     1024
#define ROWS       (BATCH*SEQLEN)   // 8192
#define LN_EPS     1e-12f
#define RMS_EPS    1e-5f

typedef __attribute__((ext_vector_type(2))) float v2f;
typedef __attribute__((ext_vector_type(8))) float v8f;

__device__ __forceinline__ float silu_f(float x) { return x / (1.0f + __expf(-x)); }

// ---------------- LayerNorm: one wave (32 lanes) per 384-wide row ----------------
__global__ void ln_kernel(const float* __restrict__ h, const float* __restrict__ w,
                          const float* __restrict__ b, float* __restrict__ hn) {
    int wave = threadIdx.x >> 5;
    int lane = threadIdx.x & 31;
    int row  = blockIdx.x * 8 + wave;          // 8192 rows, grid=1024, block=256
    const float* x = h + (size_t)row * D_MODEL;
    float v[12];
    float sum = 0.f;
#pragma unroll
    for (int j = 0; j < 12; ++j) { v[j] = x[lane + j * 32]; sum += v[j]; }
#pragma unroll
    for (int m = 16; m >= 1; m >>= 1) sum += __shfl_xor(sum, m);
    float mu = sum * (1.0f / D_MODEL);
    float var = 0.f;
#pragma unroll
    for (int j = 0; j < 12; ++j) { float d = v[j] - mu; var += d * d; }
#pragma unroll
    for (int m = 16; m >= 1; m >>= 1) var += __shfl_xor(var, m);
    float inv = rsqrtf(var * (1.0f / D_MODEL) + LN_EPS);
    float* o = hn + (size_t)row * D_MODEL;
#pragma unroll
    for (int j = 0; j < 12; ++j) {
        int c = lane + j * 32;
        o[c] = (v[j] - mu) * inv * w[c] + b[c];
    }
}

// ---------------- fp32 WMMA GEMM: C[M,N] (+)= A[M,K] * W[K,N] ----------------
// One warp computes a 32x32 block (2x2 WMMA tiles). Loads are clamp-guarded
// (no exec-mask divergence in the K-loop); stores are guarded per column.
__global__ void wmma_gemm_kernel(const float* __restrict__ Amat,
                                 const float* __restrict__ Wmat,
                                 float* __restrict__ Cmat,
                                 int N, int K, int addTo) {
    const int lane = threadIdx.x & 31;
    const int warp = threadIdx.x >> 5;
    const int mt   = (blockIdx.y * 8 + warp) * 2;   // two M-tiles: mt, mt+1
    const int nt   = blockIdx.x * 2;                // two N-tiles: nt, nt+1

    const int l16  = lane & 15;
    const int kgrp = (lane >> 4) << 1;              // half-wave K offset {0,2}

    const int mrow0 = mt * 16 + l16;
    const int mrow1 = mrow0 + 16;
    const int n0    = nt * 16 + l16;
    const int n1    = n0 + 16;
    const int n0c   = (n0 < N) ? n0 : (N - 1);      // clamp: loads always in-bounds
    const int n1c   = (n1 < N) ? n1 : (N - 1);

    const float* Ap0 = Amat + mrow0 * K + kgrp;
    const float* Ap1 = Amat + mrow1 * K + kgrp;
    const float* Wp0 = Wmat + kgrp * N + n0c;
    const float* Wp1 = Wmat + kgrp * N + n1c;
    const int    wstep = 4 * N;

    v8f acc00 = {}, acc01 = {}, acc10 = {}, acc11 = {};
#pragma unroll 4
    for (int k0 = 0; k0 < K; k0 += 4) {
        v2f a0 = *(const v2f*)Ap0;                  // global_load_b64
        v2f a1 = *(const v2f*)Ap1;
        v2f b0; b0.x = Wp0[0]; b0.y = Wp0[N];
        v2f b1; b1.x = Wp1[0]; b1.y = Wp1[N];
        acc00 = __builtin_amdgcn_wmma_f32_16x16x4_f32(false, a0, false, b0, (short)0, acc00, false, false);
        acc01 = __builtin_amdgcn_wmma_f32_16x16x4_f32(false, a0, false, b1, (short)0, acc01, false, false);
        acc10 = __builtin_amdgcn_wmma_f32_16x16x4_f32(false, a1, false, b0, (short)0, acc10, false, false);
        acc11 = __builtin_amdgcn_wmma_f32_16x16x4_f32(false, a1, false, b1, (short)0, acc11, false, false);
        Ap0 += 4; Ap1 += 4; Wp0 += wstep; Wp1 += wstep;
    }

    // store: row = tile*16 + v + 8*(lane/16), col = tile*16 + (lane&15)
    const int rsub = (lane >> 4) << 3;
    const int rb0  = mt * 16 + rsub;
    const int rb1  = rb0 + 16;
    if (n0 < N) {
#pragma unroll
        for (int v = 0; v < 8; ++v) {
            size_t o0 = (size_t)(rb0 + v) * N + n0;
            size_t o1 = (size_t)(rb1 + v) * N + n0;
            if (addTo) { Cmat[o0] += acc00[v]; Cmat[o1] += acc10[v]; }
            else       { Cmat[o0]  = acc00[v]; Cmat[o1]  = acc10[v]; }
        }
    }
    if (n1 < N) {
#pragma unroll
        for (int v = 0; v < 8; ++v) {
            size_t o0 = (size_t)(rb0 + v) * N + n1;
            size_t o1 = (size_t)(rb1 + v) * N + n1;
            if (addTo) { Cmat[o0] += acc01[v]; Cmat[o1] += acc11[v]; }
            else       { Cmat[o0]  = acc01[v]; Cmat[o1]  = acc11[v]; }
        }
    }
}

// ---------------- causal depthwise conv(4) + SiLU over xBC ----------------
__global__ void conv_kernel(const float* __restrict__ zx,
                            const float* __restrict__ cw,
                            const float* __restrict__ cb,
                            float* __restrict__ xc) {
    size_t i = (size_t)blockIdx.x * blockDim.x + threadIdx.x;
    if (i >= (size_t)ROWS * CONV_DIM) return;
    int c   = (int)(i % CONV_DIM);
    size_t bl = i / CONV_DIM;
    int l   = (int)(bl % SEQLEN);
    size_t brow0 = (bl - l) * D_IN_PROJ;       // start of this batch's rows
    const float* base = zx + brow0 + D_INNER + c;
    float acc = cb[c];
#pragma unroll
    for (int j = 0; j < D_CONV; ++j) {
        int lj = l - (D_CONV - 1) + j;
        if (lj >= 0) acc = fmaf(base[(size_t)lj * D_IN_PROJ], cw[c * D_CONV + j], acc);
    }
    xc[i] = silu_f(acc);
}

// ---------------- SSD scan: one block per (b, head), 96 threads ----------------
#define CHUNK 8
__global__ void scan_kernel(const float* __restrict__ zx,   // for dt raw
                            const float* __restrict__ xc,   // conv output [B,L,896]
                            const float* __restrict__ dt_bias,
                            const float* __restrict__ A_log,
                            const float* __restrict__ D_skip,
                            float* __restrict__ ybuf) {     // [B,L,768]
    int bh = blockIdx.x;
    int b  = bh >> 3;
    int h  = bh & 7;
    int tid = threadIdx.x;                     // 0..95 == p index

    __shared__ float s[CHUNK][224];            // per row: x[96] | B[64] | C[64]
    __shared__ float sdt[CHUNK], sdA[CHUNK];

    float A   = -__expf(A_log[h]);
    float dtb = dt_bias[h];
    float dsk = D_skip[h];

    float hst[D_STATE];
#pragma unroll
    for (int n = 0; n < D_STATE; ++n) hst[n] = 0.f;

    const float* xcb = xc + (size_t)b * SEQLEN * CONV_DIM;
    const float* zxb = zx + (size_t)b * SEQLEN * D_IN_PROJ;
    float* yb = ybuf + (size_t)b * SEQLEN * D_INNER + h * HEADDIM;

    for (int t0 = 0; t0 < SEQLEN; t0 += CHUNK) {
        // cooperative stage of CHUNK timesteps: x(head cols), B, C
        for (int idx = tid; idx < CHUNK * 224; idx += 96) {
            int r = idx / 224, c = idx % 224;
            int col = (c < HEADDIM) ? (h * HEADDIM + c) : (672 + c);  // 672+c -> 768..895
            s[r][c] = xcb[(size_t)(t0 + r) * CONV_DIM + col];
        }
        if (tid < CHUNK) {
            float raw = zxb[(size_t)(t0 + tid) * D_IN_PROJ + DT_OFF + h] + dtb;
            float dt  = (raw > 20.f) ? raw : log1pf(__expf(raw));     // softplus
            sdt[tid] = dt;
            sdA[tid] = __expf(dt * A);
        }
        __syncthreads();
        for (int r = 0; r < CHUNK; ++r) {
            float xp  = s[r][tid];
            float dtx = sdt[r] * xp;
            float dA  = sdA[r];
            const float* sB = &s[r][HEADDIM];
            const float* sC = &s[r][HEADDIM + D_STATE];
            float acc = 0.f;
#pragma unroll
            for (int n = 0; n < D_STATE; ++n) {
                hst[n] = fmaf(dA, hst[n], dtx * sB[n]);
                acc    = fmaf(hst[n], sC[n], acc);
            }
            yb[(size_t)(t0 + r) * D_INNER + tid] = acc + dsk * xp;    // fold D-skip
        }
        __syncthreads();
    }
}

// ---------------- gate (y * silu(z)) + RMSNorm, one wave per 768-wide row ----------------
__global__ void gate_rms_kernel(const float* __restrict__ zx,
                                const float* __restrict__ norm_w,
                                float* __restrict__ ybuf) {
    int wave = threadIdx.x >> 5;
    int lane = threadIdx.x & 31;
    int row  = blockIdx.x * 8 + wave;          // 8192 rows
    const float* z = zx + (size_t)row * D_IN_PROJ;   // z = first 768 cols
    float* y = ybuf + (size_t)row * D_INNER;
    float g[24];
    float ss = 0.f;
#pragma unroll
    for (int j = 0; j < 24; ++j) {
        int c = lane + j * 32;
        float gv = y[c] * silu_f(z[c]);
        g[j] = gv;
        ss += gv * gv;
    }
#pragma unroll
    for (int m = 16; m >= 1; m >>= 1) ss += __shfl_xor(ss, m);
    float scale = rsqrtf(ss * (1.0f / D_INNER) + RMS_EPS);
#pragma unroll
    for (int j = 0; j < 24; ++j) {
        int c = lane + j * 32;
        y[c] = g[j] * scale * norm_w[c];
    }
}

// ---------------- launcher ----------------
extern "C" void kernel_launch(void* const* d_in, const int* in_sizes, int n_in,
                              void* d_out, int out_size, void* d_ws, size_t ws_size,
                              hipStream_t stream) {
    const float* x_in    = (const float*)d_in[0];
    const float* ln_w    = (const float*)d_in[1];
    const float* ln_b    = (const float*)d_in[2];
    const float* W_in    = (const float*)d_in[3];   // [6, 384, 1672]
    const float* conv_w  = (const float*)d_in[4];   // [6, 896, 4]
    const float* conv_b  = (const float*)d_in[5];   // [6, 896]
    const float* dt_bias = (const float*)d_in[6];   // [6, 8]
    const float* A_log   = (const float*)d_in[7];   // [6, 8]
    const float* D_skip  = (const float*)d_in[8];   // [6, 8]
    const float* norm_w  = (const float*)d_in[9];   // [6, 768]
    const float* W_out   = (const float*)d_in[10];  // [6, 768, 384]

    float* h = (float*)d_out;                       // running hidden state [8,1024,384]

    // workspace carve-up
    char* ws = (char*)d_ws;
    float* hn   = (float*)ws;                                 ws += (size_t)ROWS * D_MODEL   * sizeof(float);
    float* zx   = (float*)ws;                                 ws += (size_t)ROWS * D_IN_PROJ * sizeof(float);
    float* xc   = (float*)ws;                                 ws += (size_t)ROWS * CONV_DIM  * sizeof(float);
    float* ybuf = (float*)ws;

    // h = input embedding
    hipMemcpyAsync(h, x_in, (size_t)ROWS * D_MODEL * sizeof(float),
                   hipMemcpyDeviceToDevice, stream);

    const int MT = ROWS / 16;                       // 512 M-tiles -> 256 tile-pairs
    dim3 gemmBlk(256);                              // 8 warps; each warp = 2x2 tiles
    dim3 gridIn((D_IN_PROJ / 16 + 2) / 2, MT / 16); // 53 x 32  (covers 106 N-tiles)
    dim3 gridOut((D_MODEL / 16) / 2,     MT / 16);  // 12 x 32

    for (int i = 0; i < NUM_LAYERS; ++i) {
        const float* Wi  = W_in   + (size_t)i * D_MODEL * D_IN_PROJ;
        const float* cwi = conv_w + (size_t)i * CONV_DIM * D_CONV;
        const float* cbi = conv_b + (size_t)i * CONV_DIM;
        const float* Wo  = W_out  + (size_t)i * D_INNER * D_MODEL;

        ln_kernel<<<ROWS / 8, 256, 0, stream>>>(h, ln_w + i * D_MODEL, ln_b + i * D_MODEL, hn);

        wmma_gemm_kernel<<<gridIn, gemmBlk, 0, stream>>>(hn, Wi, zx, D_IN_PROJ, D_MODEL, 0);

        {
            size_t total = (size_t)ROWS * CONV_DIM;
            int blocks = (int)((total + 255) / 256);
            conv_kernel<<<blocks, 256, 0, stream>>>(zx, cwi, cbi, xc);
        }

        scan_kernel<<<BATCH * NHEADS, 96, 0, stream>>>(
            zx, xc, dt_bias + i * NHEADS, A_log + i * NHEADS, D_skip + i * NHEADS, ybuf);

        gate_rms_kernel<<<ROWS / 8, 256, 0, stream>>>(zx, norm_w + i * D_INNER, ybuf);

        wmma_gemm_kernel<<<gridOut, gemmBlk, 0, stream>>>(ybuf, Wo, h, D_MODEL, D_INNER, 1);
    }
}